// MSDeformAttn_11484742549699
// MI455X (gfx1250) — compile-verified
//
#include <hip/hip_runtime.h>
#include <hip/hip_bf16.h>
#include <math.h>

// ---- problem constants (fixed by the reference module) ----
#define D_MODEL   256
#define N_HEADS   8
#define N_LEVELS  4
#define N_POINTS  4
#define B_SZ      2
#define LQ_SZ     21760          // 128^2 + 64^2 + 32^2 + 16^2
#define DH        32             // D_MODEL / N_HEADS
#define M_ROWS    (B_SZ * LQ_SZ) // 43520, multiple of 16
#define KDIM      256            // all GEMMs share K = 256
#define LDSK      260            // padded LDS row stride (floats): 260%64=4 -> conflict-free

typedef __attribute__((ext_vector_type(2))) float v2f;
typedef __attribute__((ext_vector_type(8))) float v8f;
typedef __attribute__((ext_vector_type(4))) int   v4i_t;

#define AS1 __attribute__((address_space(1)))
#define AS3 __attribute__((address_space(3)))

static __device__ __forceinline__ v8f wmma_f32(v2f a, v2f b, v8f c) {
  return __builtin_amdgcn_wmma_f32_16x16x4_f32(
      /*neg_a=*/false, a, /*neg_b=*/false, b,
      /*c_mod=*/(short)0, c, /*reuse_a=*/false, /*reuse_b=*/false);
}

// ---------------------------------------------------------------------------
// fp32 WMMA GEMM:  C[M,N] = A[M,KDIM] * B[KDIM,N] + bias[N],  N in {128, 256}
// One block = one 16-row M panel covering all N.  The 16x256 A panel is
// staged into LDS (async global->LDS copy when available), then each of the
// 8 waves computes one or two 16x16 tiles with V_WMMA_F32_16X16X4_F32,
// sharing the A fragment between two independent accumulator chains.
// A-operand layout (16x4 f32, 2 VGPRs): lane&15 = M-row; lanes 0-15 hold
// K=k,k+1, lanes 16-31 hold K=k+2,k+3.  B-operand (4x16) mirrored with
// lane&15 = N-col.  C/D: VGPR i holds rows (i) [lanes 0-15] / (i+8) [16-31].
// ---------------------------------------------------------------------------
__global__ __launch_bounds__(256)
void wmma_gemm_bias_f32(const float* __restrict__ A,
                        const float* __restrict__ Bm,
                        const float* __restrict__ bias,
                        float* __restrict__ C,
                        int N) {
  __shared__ float As[16 * LDSK];

  const int tid  = threadIdx.x;
  const int lane = tid & 31;
  const int wave = tid >> 5;
  const int tm   = blockIdx.x;              // M tile row

  // ---- stage A[tm*16 .. +16][0..256) into LDS (padded rows) ----
  const float* Ablk = A + (size_t)tm * 16 * KDIM;
  // 1024 x b128 chunks; 256 threads -> 4 chunks each
#if __has_builtin(__builtin_amdgcn_global_load_async_to_lds_b128)
  for (int i = tid; i < 1024; i += 256) {
    const int row = i >> 6;                 // 64 chunks per row
    const int off = (i & 63) * 4;           // float offset within row
    __builtin_amdgcn_global_load_async_to_lds_b128(
        (AS1 v4i_t*)(Ablk + (size_t)row * KDIM + off),
        (AS3 v4i_t*)(As + row * LDSK + off),
        /*offset=*/0, /*cpol=*/0);
  }
  __builtin_amdgcn_s_wait_asynccnt(0);
#else
  for (int i = tid; i < 1024; i += 256) {
    const int row = i >> 6;
    const int off = (i & 63) * 4;
    const float4 v = *(const float4*)(Ablk + (size_t)row * KDIM + off);
    *(float4*)(As + row * LDSK + off) = v;
  }
#endif
  __syncthreads();

  const int ntn   = N >> 4;                 // 8 or 16 tiles along N
  const int mrow  = lane & 15;
  const int khalf = (lane >> 4) << 1;       // 0 for lanes 0-15, 2 for 16-31
  const float* arow = As + mrow * LDSK;

  const int col0 = wave * 16 + (lane & 15);        // tile tn = wave
  const int col1 = (wave + 8) * 16 + (lane & 15);  // tile tn = wave + 8

  v8f acc0 = {};
  v8f acc1 = {};

  if (ntn > 8) {
    for (int k = 0; k < KDIM; k += 4) {
      v2f a, b0, b1;
      a.x  = arow[k + khalf + 0];
      a.y  = arow[k + khalf + 1];
      b0.x = Bm[(size_t)(k + khalf + 0) * N + col0];
      b0.y = Bm[(size_t)(k + khalf + 1) * N + col0];
      b1.x = Bm[(size_t)(k + khalf + 0) * N + col1];
      b1.y = Bm[(size_t)(k + khalf + 1) * N + col1];
      acc0 = wmma_f32(a, b0, acc0);
      acc1 = wmma_f32(a, b1, acc1);
    }
  } else {
    for (int k = 0; k < KDIM; k += 4) {
      v2f a, b0;
      a.x  = arow[k + khalf + 0];
      a.y  = arow[k + khalf + 1];
      b0.x = Bm[(size_t)(k + khalf + 0) * N + col0];
      b0.y = Bm[(size_t)(k + khalf + 1) * N + col0];
      acc0 = wmma_f32(a, b0, acc0);
    }
  }

  // ---- write back with bias ----
  const int rbase = tm * 16 + ((lane >> 4) << 3);   // +0 or +8
  {
    const float bv = bias[col0];
#pragma unroll
    for (int i = 0; i < 8; ++i)
      C[(size_t)(rbase + i) * N + col0] = acc0[i] + bv;
  }
  if (ntn > 8) {
    const float bv = bias[col1];
#pragma unroll
    for (int i = 0; i < 8; ++i)
      C[(size_t)(rbase + i) * N + col1] = acc1[i] + bv;
  }
}

// ---------------------------------------------------------------------------
// Softmax over the 16 (level,point) logits per (b, q, head).  In-place.
// ---------------------------------------------------------------------------
__global__ __launch_bounds__(256)
void softmax16_kernel(float* __restrict__ aw, int total) {
  const int i = blockIdx.x * blockDim.x + threadIdx.x;
  if (i >= total) return;
  float* p = aw + (size_t)i * 16;
  float v[16];
  float m = -INFINITY;
#pragma unroll
  for (int j = 0; j < 16; ++j) { v[j] = p[j]; m = fmaxf(m, v[j]); }
  float s = 0.f;
#pragma unroll
  for (int j = 0; j < 16; ++j) { v[j] = __expf(v[j] - m); s += v[j]; }
  const float inv = 1.f / s;
#pragma unroll
  for (int j = 0; j < 16; ++j) p[j] = v[j] * inv;
}

// ---------------------------------------------------------------------------
// Deformable sampling core.  Thread = (b, q, head, 8-channel chunk).
// core[b,q,h*32+c*8 .. +8] = sum_{lvl,pt} aw * bilinear(vproj)
// ---------------------------------------------------------------------------
__global__ __launch_bounds__(256)
void ms_deform_sample_kernel(const float* __restrict__ ref,    // [B,LQ,4,2]
                             const float* __restrict__ off,    // [B,LQ,256]
                             const float* __restrict__ aw,     // [B,LQ,8,16]
                             const float* __restrict__ vproj,  // [B,LQ,256] as [.,.,h,32]
                             float* __restrict__ core) {       // [B,LQ,256]
  const int total = B_SZ * LQ_SZ * N_HEADS * 4;
  const int t = blockIdx.x * blockDim.x + threadIdx.x;
  if (t >= total) return;

  const int c  = t & 3;                 // channel chunk (8 floats)
  const int h  = (t >> 2) & 7;          // head
  const int bq = t >> 5;                // b*LQ + q
  const int ch = h * DH + c * 8;

  const int Hs[4]  = {128, 64, 32, 16};
  const int Lst[4] = {0, 16384, 20480, 21504};

  float acc[8];
#pragma unroll
  for (int j = 0; j < 8; ++j) acc[j] = 0.f;

  const float* refq = ref + (size_t)bq * (N_LEVELS * 2);
  const float* offq = off + (size_t)bq * D_MODEL + (size_t)h * (N_LEVELS * N_POINTS * 2);
  const float* awq  = aw  + (size_t)bq * (N_HEADS * 16) + (size_t)h * 16;
  const int b = bq / LQ_SZ;

#pragma unroll
  for (int lvl = 0; lvl < N_LEVELS; ++lvl) {
    const int   Wi = Hs[lvl];           // square levels: W == H
    const int   Hi = Hs[lvl];
    const float Wf = (float)Wi, Hf = (float)Hi;
    const float rx = refq[lvl * 2 + 0];
    const float ry = refq[lvl * 2 + 1];
    const float* vbase = vproj + ((size_t)(b * LQ_SZ + Lst[lvl])) * D_MODEL + ch;

#pragma unroll
    for (int p = 0; p < N_POINTS; ++p) {
      const float ox = offq[(lvl * N_POINTS + p) * 2 + 0];
      const float oy = offq[(lvl * N_POINTS + p) * 2 + 1];
      const float awv = awq[lvl * N_POINTS + p];

      const float x = (rx + ox / Wf) * Wf - 0.5f;
      const float y = (ry + oy / Hf) * Hf - 0.5f;
      const float x0f = floorf(x), y0f = floorf(y);
      const float fx = x - x0f, fy = y - y0f;
      const int x0 = (int)x0f, y0 = (int)y0f;

      const float cw[4] = {(1.f - fx) * (1.f - fy), fx * (1.f - fy),
                           (1.f - fx) * fy,          fx * fy};
      const int dxs[4] = {0, 1, 0, 1};
      const int dys[4] = {0, 0, 1, 1};

#pragma unroll
      for (int cc = 0; cc < 4; ++cc) {
        const int xi = x0 + dxs[cc];
        const int yi = y0 + dys[cc];
        const bool valid = (xi >= 0) & (xi < Wi) & (yi >= 0) & (yi < Hi);
        const int xc = xi < 0 ? 0 : (xi > Wi - 1 ? Wi - 1 : xi);
        const int yc = yi < 0 ? 0 : (yi > Hi - 1 ? Hi - 1 : yi);
        const float wt = valid ? cw[cc] * awv : 0.f;
        const float4* g = (const float4*)(vbase + (size_t)(yc * Wi + xc) * D_MODEL);
        const float4 g0 = g[0];
        const float4 g1 = g[1];
        acc[0] = fmaf(g0.x, wt, acc[0]);
        acc[1] = fmaf(g0.y, wt, acc[1]);
        acc[2] = fmaf(g0.z, wt, acc[2]);
        acc[3] = fmaf(g0.w, wt, acc[3]);
        acc[4] = fmaf(g1.x, wt, acc[4]);
        acc[5] = fmaf(g1.y, wt, acc[5]);
        acc[6] = fmaf(g1.z, wt, acc[6]);
        acc[7] = fmaf(g1.w, wt, acc[7]);
      }
    }
  }

  float* o = core + (size_t)bq * D_MODEL + ch;
#pragma unroll
  for (int j = 0; j < 8; ++j) o[j] = acc[j];
}

// ---------------------------------------------------------------------------
extern "C" void kernel_launch(void* const* d_in, const int* in_sizes, int n_in,
                              void* d_out, int out_size, void* d_ws, size_t ws_size,
                              hipStream_t stream) {
  const float* query   = (const float*)d_in[0];   // [B,LQ,256]
  const float* refpts  = (const float*)d_in[1];   // [B,LQ,4,2]
  const float* value   = (const float*)d_in[2];   // [B,LQ,256]
  // d_in[3] spatial_shapes, d_in[4] level_start_index: compile-time constants
  const float* W_value = (const float*)d_in[5];   // [256,256]
  const float* b_value = (const float*)d_in[6];   // [256]
  const float* W_off   = (const float*)d_in[7];   // [256,256]
  const float* b_off   = (const float*)d_in[8];   // [256]
  const float* W_attn  = (const float*)d_in[9];   // [256,128]
  const float* b_attn  = (const float*)d_in[10];  // [128]
  const float* W_out   = (const float*)d_in[11];  // [256,256]
  const float* b_out   = (const float*)d_in[12];  // [256]
  float* out = (float*)d_out;

  // workspace layout (floats)
  float* ws    = (float*)d_ws;
  float* vproj = ws;                               // 43520*256
  float* off   = vproj + (size_t)M_ROWS * 256;     // 43520*256
  float* aw    = off   + (size_t)M_ROWS * 256;     // 43520*128
  float* core  = aw    + (size_t)M_ROWS * 128;     // 43520*256

  const int mblocks = M_ROWS / 16;                 // 2720

  // 1) v = value @ W_value + b_value   (N=256)
  wmma_gemm_bias_f32<<<mblocks, 256, 0, stream>>>(value, W_value, b_value, vproj, 256);
  // 2) off = query @ W_offset + b_offset  (N=256)
  wmma_gemm_bias_f32<<<mblocks, 256, 0, stream>>>(query, W_off, b_off, off, 256);
  // 3) aw_logits = query @ W_attn + b_attn  (N=128)
  wmma_gemm_bias_f32<<<mblocks, 256, 0, stream>>>(query, W_attn, b_attn, aw, 128);
  // 4) softmax over 16 logits per (b,q,head)
  {
    int total = M_ROWS * N_HEADS;
    int blocks = (total + 255) / 256;
    softmax16_kernel<<<blocks, 256, 0, stream>>>(aw, total);
  }
  // 5) deformable bilinear sampling
  {
    int total = M_ROWS * N_HEADS * 4;
    int blocks = (total + 255) / 256;
    ms_deform_sample_kernel<<<blocks, 256, 0, stream>>>(refpts, off, aw, vproj, core);
  }
  // 6) out = core @ W_out + b_out  (N=256) -> d_out
  wmma_gemm_bias_f32<<<mblocks, 256, 0, stream>>>(core, W_out, b_out, out, 256);
}